// CachedShapingFunctions_10342281249362
// MI455X (gfx1250) — compile-verified
//
#include <hip/hip_runtime.h>
#include <hip/hip_bf16.h>

// ---------------------------------------------------------------------------
// CachedShapingFunctions: out[b,t,w] = lerp of lookup_table[:,w] at
// ic = (NB-1)*(x - MIN)/(MAX-MIN).  Memory-bound: 268MB HBM traffic
// (~11.5us floor @ 23.3 TB/s).  Strategy: stage a 4-column LDS tile of the
// LUT per block (padded stride 5 dwords -> gcd(5,64)=1 -> random gathers
// spread over all 64 banks), staged with CDNA5 async global->LDS loads
// (ASYNCcnt), then ds_load_b32 gathers + fma lerp.  Grid x = w-chunk
// (fastest varying) so the 16 co-resident chunk-blocks covering the same
// rows share input cachelines through the 192MB L2.
// ---------------------------------------------------------------------------

#define MINV (-3.0f)
#define MAXV (3.0f)

constexpr int NB     = 4096;     // buckets
constexpr int W      = 64;       // waveshapers
constexpr int BATCH  = 16;
constexpr int TLEN   = 32768;
constexpr long long NROWS = (long long)BATCH * TLEN;   // 524288 rows of 64 floats
constexpr int WC     = 4;        // LUT columns staged per block
constexpr int SP     = 5;        // padded LDS stride per bucket (dwords); gcd(5,64)=1
constexpr int NCHUNK = W / WC;   // 16
constexpr int BLOCK  = 256;      // 8 waves (wave32)
constexpr int GY     = 128;      // row tiles
constexpr int ROWS_PER_BLOCK = (int)(NROWS / GY);      // 4096

typedef float v4f __attribute__((ext_vector_type(4)));

// CDNA5 async global->LDS copy of one 16B LUT bucket row into the padded
// tile.  GLOBAL_LOAD_ASYNC_TO_LDS_B32's INST_OFFSET is added to BOTH the
// global and LDS addresses (CDNA5 ISA), so the four dword
// copies share one address pair and differ only in the immediate offset.
__device__ __forceinline__ void async_stage_row(const float* g, float* l) {
#if __has_builtin(__builtin_amdgcn_global_load_async_to_lds_b32)
    __builtin_amdgcn_global_load_async_to_lds_b32((int*)g, (int*)l, 0,  0);
    __builtin_amdgcn_global_load_async_to_lds_b32((int*)g, (int*)l, 4,  0);
    __builtin_amdgcn_global_load_async_to_lds_b32((int*)g, (int*)l, 8,  0);
    __builtin_amdgcn_global_load_async_to_lds_b32((int*)g, (int*)l, 12, 0);
#else
    l[0] = g[0]; l[1] = g[1]; l[2] = g[2]; l[3] = g[3];
#endif
}

__device__ __forceinline__ void wait_asynccnt0() {
#if __has_builtin(__builtin_amdgcn_s_wait_asynccnt)
    __builtin_amdgcn_s_wait_asynccnt(0);
#else
    asm volatile("s_wait_asynccnt 0" ::: "memory");
#endif
}

__global__ __launch_bounds__(BLOCK)
void CachedShapingFunctions_10342281249362_kernel(const float* __restrict__ in,
                                                  const float* __restrict__ lut,
                                                  float* __restrict__ out) {
    // Padded LDS tile: tile[b*SP + k] = lut[b*W + w0 + k], k in [0,WC)
    // 4096 * 5 * 4B = 80 KB -> up to 4 blocks per 320KB WGP
    __shared__ float tile[NB * SP];

    const int tid  = threadIdx.x;
    const int w0   = blockIdx.x * WC;              // chunk id varies fastest
    const int row0 = blockIdx.y * ROWS_PER_BLOCK;

    // ---- stage LUT columns w0..w0+3 into LDS via async copies ----
    for (int b = tid; b < NB; b += BLOCK) {
        async_stage_row(lut + (size_t)b * W + w0, tile + b * SP);
    }
    wait_asynccnt0();
    __syncthreads();

    // ---- streaming lerp: 16 row-iterations/thread, unrolled x4 for MLP ----
    const float scale = (float)(NB - 1) / (MAXV - MINV);
    const int rend = row0 + ROWS_PER_BLOCK;
#pragma unroll 4
    for (int r = row0 + tid; r < rend; r += BLOCK) {
        const size_t base = (size_t)r * W + w0;
        v4f x = *(const v4f*)(in + base);
        v4f y;
#pragma unroll
        for (int k = 0; k < 4; ++k) {
            float ic = (x[k] - MINV) * scale;
            int   il = (int)floorf(ic);
            il = il < 0 ? 0 : (il > NB - 1 ? NB - 1 : il);
            int   iu = il + 1 > NB - 1 ? NB - 1 : il + 1;
            float fr = ic - (float)il;         // matches reference (clamped il)
            float lo = tile[il * SP + k];      // ds_load_b32 gather
            float hi = tile[iu * SP + k];      // banks = (5*il[+5]+k)%64: full spread
            y[k] = fmaf(hi - lo, fr, lo);
        }
        *(v4f*)(out + base) = y;
    }
}

extern "C" void kernel_launch(void* const* d_in, const int* in_sizes, int n_in,
                              void* d_out, int out_size, void* d_ws, size_t ws_size,
                              hipStream_t stream) {
    const float* in  = (const float*)d_in[0];   // [16, 32768, 64] f32
    const float* lut = (const float*)d_in[1];   // [4096, 64] f32
    float*       out = (float*)d_out;           // [16, 32768, 64] f32

    dim3 grid(NCHUNK, GY);                      // x = w-chunk fastest -> L2 reuse
    CachedShapingFunctions_10342281249362_kernel<<<grid, BLOCK, 0, stream>>>(in, lut, out);
}